// PositionalEncoding_24017457119355
// MI455X (gfx1250) — compile-verified
//
#include <hip/hip_runtime.h>
#include <stdint.h>

// ---------------------------------------------------------------------------
// PositionalEncoding expansion for run-length tokens.
//   rows B=32, tokens L=512, embed D=384 (= 96 float4), token len in [0,16)
// Memory-bound: ~200MB NT writes (HBM floor ~8.6us @ 23.3TB/s); pos_enc
// (7.3MB) stays L2-resident. CDNA5 paths used: global_load_async_to_lds_b128
// (ASYNCcnt) for the per-row search table, wave32 shuffle scan, NT B128 stores.
// ---------------------------------------------------------------------------

#define B_ROWS 32
#define L_TOK  512
#define D_EMB  384
#define D_VEC4 (D_EMB / 4)   // 96 float4 per row

typedef float v4f __attribute__((ext_vector_type(4)));

// ---------------------------------------------------------------------------
// Kernel 1: per-row inclusive cumsum + starts + totals.
// One wave (32 lanes) per row; lane l handles tokens [16l, 16l+16).
// ws row block (4KB, contiguous for one async B128 stage in kernel 2):
//   ints [0..511]   = cs      (inclusive cumsum)
//   ints [512..1023]= starts  (cs - x = exclusive cumsum)
// ---------------------------------------------------------------------------
__global__ void pe_cumsum_kernel(const int* __restrict__ x,
                                 int* __restrict__ ws_rows,   // 32 * 1024 ints
                                 int* __restrict__ totals)    // 32 ints
{
    const int row  = blockIdx.x;
    const int lane = threadIdx.x;            // 0..31 (wave32)

    const int* xr = x + row * L_TOK + lane * 16;

    int vals[16];
    int lsum = 0;
#pragma unroll
    for (int i = 0; i < 16; ++i) { vals[i] = xr[i]; lsum += vals[i]; }

    // wave32 inclusive scan of per-lane sums -> exclusive offset
    int incl = lsum;
#pragma unroll
    for (int d = 1; d < 32; d <<= 1) {
        int n = __shfl_up(incl, d, 32);
        if (lane >= d) incl += n;
    }
    int run = incl - lsum;                   // exclusive prefix for this lane

    int* rowws = ws_rows + row * 1024;
#pragma unroll
    for (int i = 0; i < 16; ++i) {
        int st = run;                        // starts = cumsum before this token
        run += vals[i];
        rowws[lane * 16 + i]       = run;    // cs (inclusive)
        rowws[512 + lane * 16 + i] = st;     // starts
    }
    if (lane == 31) totals[row] = run;       // cs[511]
}

// ---------------------------------------------------------------------------
// Kernel 2: expand. Block = 256 threads = 8 waves; wave w handles timestep
// t = blockIdx.x*8 + w for row blockIdx.y. The row's 4KB (cs+starts) table is
// staged into LDS with per-lane async B128 copies (CDNA5 ASYNCcnt path), then
// each wave does a 9-step upper_bound and streams one 1536B pos_enc row to
// the output with non-temporal B128 stores (keeps pos_enc hot in L2).
// ---------------------------------------------------------------------------
__global__ void pe_expand_kernel(const float* __restrict__ pos_enc,
                                 const int* __restrict__ ws_rows,
                                 const int* __restrict__ totals,
                                 float* __restrict__ out,
                                 int T)
{
    __shared__ int smem[1024];               // [0..511]=cs  [512..1023]=starts

    const int tid = threadIdx.x;             // 0..255
    const int row = blockIdx.y;

    // ---- async stage: 256 lanes x 16B = 4KB  (global -> LDS, ASYNCcnt) ----
    {
        const int* src = ws_rows + row * 1024;           // SADDR (64-bit SGPR pair)
        unsigned voff  = (unsigned)(tid * 16);           // GVS: saddr + vaddr32
        // low 32 bits of a flat shared pointer == LDS byte offset (aperture
        // base lives in addr[63:32] on CDNA5)
        unsigned ldsa  = (unsigned)(uintptr_t)(&smem[0]) + (unsigned)(tid * 16);
        asm volatile(
            "global_load_async_to_lds_b128 %0, %1, %2\n\t"
            "s_wait_asynccnt 0"
            :
            : "v"(ldsa), "v"(voff), "s"(src)
            : "memory");
    }
    __syncthreads();

    const int wave = tid >> 5;
    const int lane = tid & 31;
    const int t    = blockIdx.x * 8 + wave;
    if (t >= T) return;

    const int total = totals[row];
    v4f* __restrict__ dst =
        (v4f*)out + ((size_t)row * (size_t)T + (size_t)t) * D_VEC4;

    if (t >= total) {
        v4f z = (v4f)0.0f;
#pragma unroll
        for (int j = 0; j < 3; ++j)
            __builtin_nontemporal_store(z, &dst[lane + 32 * j]);
        return;
    }

    // upper_bound over cs[512]: idx = #elements <= t  (searchsorted 'right')
    int idx = 0;
#pragma unroll
    for (int s = 256; s >= 1; s >>= 1) {
        int c = idx + s;
        if (c <= L_TOK && smem[c - 1] <= t) idx = c;
    }
    const int tok = (idx < L_TOK - 1) ? idx : (L_TOK - 1);
    const int pos = t - smem[512 + tok];     // in [0, 16)

    const v4f* __restrict__ src4 =
        (const v4f*)(pos_enc + (size_t)pos * D_EMB);
#pragma unroll
    for (int j = 0; j < 3; ++j) {
        v4f v = src4[lane + 32 * j];                      // L2-resident gather
        __builtin_nontemporal_store(v, &dst[lane + 32 * j]);  // NT B128 store
    }
}

// ---------------------------------------------------------------------------
extern "C" void kernel_launch(void* const* d_in, const int* in_sizes, int n_in,
                              void* d_out, int out_size, void* d_ws, size_t ws_size,
                              hipStream_t stream)
{
    const int*   x       = (const int*)d_in[0];     // (32, 512) int32
    const float* pos_enc = (const float*)d_in[1];   // (5000, 384) float32
    float*       out     = (float*)d_out;           // (32, T, 384) float32

    const int T = out_size / (B_ROWS * D_EMB);

    int* ws_rows = (int*)d_ws;                      // 32 * 1024 ints = 128KB
    int* totals  = ws_rows + B_ROWS * 1024;         // + 32 ints

    pe_cumsum_kernel<<<B_ROWS, 32, 0, stream>>>(x, ws_rows, totals);

    int tblocks = (T + 7) / 8;
    if (tblocks < 1) tblocks = 1;
    dim3 grid(tblocks, B_ROWS);
    pe_expand_kernel<<<grid, 256, 0, stream>>>(pos_enc, ws_rows, totals, out, T);
}